// MixedDiffusion_58669253263703
// MI455X (gfx1250) — compile-verified
//
#include <hip/hip_runtime.h>
#include <hip/hip_bf16.h>
#include <math.h>

#define T_STEPS 1000
#define NCAT 16
#define KCLS 16
#define SUMK 256
#define BATCH 64
#define LSEQ 256
#define NUM_N (BATCH * 8 * LSEQ)   // 131072

typedef float v2f __attribute__((ext_vector_type(2)));
typedef float v8f __attribute__((ext_vector_type(8)));

__device__ __forceinline__ float lae(float a, float b) {
    float m = fmaxf(a, b);
    return m + logf(expf(a - m) + expf(b - m));
}

// Exact f32 sum of all 32 lanes of the wave using two chained
// v_wmma_f32_16x16x4_f32 against an all-ones B matrix.
// A layout (16x4 f32): lanes 0-15 vgpr0 = A[m][0], lanes 16-31 vgpr0 = A[m][2];
// vgpr1 (K=1,3) = 0.  => D[m][n] = v_m + v_{m+16}.
// Per-lane sum of the 8 D vgprs gives half-sums; second WMMA adds the halves
// and broadcasts the total to every lane. Requires EXEC == all ones.
__device__ __forceinline__ float wave_sum_wmma(float v) {
    v2f a;  a[0] = v;    a[1] = 0.0f;
    v2f one; one[0] = 1.0f; one[1] = 1.0f;
    v8f c = {};
    v8f d = __builtin_amdgcn_wmma_f32_16x16x4_f32(false, a, false, one,
                                                  (short)0, c, false, false);
    float s = ((d[0] + d[1]) + (d[2] + d[3])) + ((d[4] + d[5]) + (d[6] + d[7]));
    v2f a2; a2[0] = s; a2[1] = 0.0f;
    v8f d2 = __builtin_amdgcn_wmma_f32_16x16x4_f32(false, a2, false, one,
                                                   (short)0, c, false, false);
    return d2[0];
}

// ---------------------------------------------------------------------------
// Kernel 1: zero accumulators + cosine beta schedule buffers (f64 -> f32),
// matching numpy float64 math in the reference.
// ---------------------------------------------------------------------------
__global__ void md_sched_kernel(float* la, float* l1a, float* lc, float* l1c,
                                double* acc) {
    if (threadIdx.x != 0 || blockIdx.x != 0) return;
    acc[0] = 0.0;   // categorical Lt sum
    acc[1] = 0.0;   // numeric squared-error sum
    const double s = 0.008;
    const double half_pi = 1.5707963267948966;
    double prev = cos((s / (1.0 + s)) * half_pi); prev *= prev;   // ac(0)
    double cum = 0.0;
    for (int i = 0; i < T_STEPS; ++i) {
        double xx = (double)(i + 1) / (double)T_STEPS;
        double c = cos(((xx + s) / (1.0 + s)) * half_pi); c *= c; // ac(i+1)
        double beta = 1.0 - c / prev;
        if (beta < 0.0)   beta = 0.0;
        if (beta > 0.999) beta = 0.999;
        prev = c;
        double log_a = log(1.0 - beta);
        cum += log_a;
        la[i]  = (float)log_a;
        l1a[i] = (float)log(1.0 - exp(log_a) + 1e-40);
        lc[i]  = (float)cum;
        l1c[i] = (float)log(1.0 - exp(cum) + 1e-40);
    }
}

// ---------------------------------------------------------------------------
// Kernel 2: categorical loss. One block per (b, n_cat); thread = position l.
// Each thread handles one K=16 class segment (class stride = LSEQ -> loads
// are fully coalesced across the 32 lanes along l). Linear-domain math keeps
// the TRANS-pipe op count near the HBM roofline.
// ---------------------------------------------------------------------------
__global__ void __launch_bounds__(LSEQ)
md_cat_kernel(const int* __restrict__ cats,
              const float* __restrict__ logits,
              const float* __restrict__ u_noise,
              const int* __restrict__ t_arr,
              const float* __restrict__ la, const float* __restrict__ l1a,
              const float* __restrict__ lc, const float* __restrict__ l1c,
              double* __restrict__ acc) {
    const int b = blockIdx.x >> 4;
    const int n = blockIdx.x & 15;
    const int l = threadIdx.x;

    const float LOG_EPS = logf(1e-30f);     // log one-hot "zero"
    const float LOG_KF  = logf(16.0f);

    const int  t  = t_arr[b];
    const bool t0 = (t == 0);
    const int  tp = t0 ? 0 : (t - 1);
    const float lc_t  = lc[t],  l1c_t = l1c[t];
    const float la_t  = la[t],  l1a_t = l1a[t];
    const float lc_p  = lc[tp], l1c_p = l1c[tp];

    // one-hot collapsed q_pred / q_pred_one values (uniform per block)
    const float ev_hit  = lae(lc_t, l1c_t - LOG_KF);
    const float ev_miss = lae(LOG_EPS + lc_t, l1c_t - LOG_KF);
    const float qx_hit  = lae(la_t, l1a_t - LOG_KF);
    const float qx_miss = lae(LOG_EPS + la_t, l1a_t - LOG_KF);
    const float e0_hit  = t0 ? 0.0f    : lae(lc_p, l1c_p - LOG_KF);
    const float e0_miss = t0 ? LOG_EPS : lae(LOG_EPS + lc_p, l1c_p - LOG_KF);
    const float d_hit   = expf(-ev_hit);      // for argmin transform
    const float d_miss  = expf(-ev_miss);
    const float Q_hit   = expf(qx_hit);
    const float Q_miss  = expf(qx_miss);
    const float exp_lc  = t0 ? 1.0f : expf(lc_p);
    const float Cc      = t0 ? 0.0f : expf(l1c_p - LOG_KF);

    const int cat = cats[(b * NCAT + n) * LSEQ + l];
    const size_t base = (size_t)(b * SUMK + n * KCLS) * LSEQ + (size_t)l;

    // ---- gumbel q_sample: argmax_k(ev_k + g_k) == argmin_k(w_k * e^{-ev_k})
    //      with w_k = -log(u_k + 1e-30) + 1e-30 (monotone transform) ----
    int samp = 0; float best = 3.0e38f;
    #pragma unroll
    for (int k = 0; k < KCLS; ++k) {
        float u = u_noise[base + (size_t)k * LSEQ];
        float w = -logf(u + 1e-30f) + 1e-30f;
        float sc = w * ((k == cat) ? d_hit : d_miss);
        if (sc < best) { best = sc; samp = k; }
    }

    // ---- softmax numerators of model logits ----
    float x[KCLS];
    float xm = -3.0e38f;
    #pragma unroll
    for (int k = 0; k < KCLS; ++k) {
        x[k] = logits[base + (size_t)k * LSEQ];
        xm = fmaxf(xm, x[k]);
    }
    float se = 0.0f, s_samp = 0.0f;
    #pragma unroll
    for (int k = 0; k < KCLS; ++k) {
        float sk = expf(x[k] - xm);
        x[k] = sk;                       // reuse registers: x[] now holds s_k
        se += sk;
        s_samp = (k == samp) ? sk : s_samp;
    }

    // ---- true posterior: un_true takes <= 4 distinct values ----
    const float ut_cc = e0_hit  + qx_hit;    // k == cat == samp
    const float ut_c  = e0_hit  + qx_miss;   // k == cat != samp
    const float ut_s  = e0_miss + qx_hit;    // k == samp != cat
    const float ut_o  = e0_miss + qx_miss;   // otherwise
    float lse_t;
    if (cat == samp) {
        float m = fmaxf(ut_cc, ut_o);
        lse_t = m + logf(expf(ut_cc - m) + 15.0f * expf(ut_o - m));
    } else {
        float m = fmaxf(fmaxf(ut_c, ut_s), ut_o);
        lse_t = m + logf(expf(ut_c - m) + expf(ut_s - m) + 14.0f * expf(ut_o - m));
    }
    // distinct exp(log_true) values
    const float pt_cat  = expf(((cat == samp) ? ut_cc : ut_c) - lse_t);
    const float pt_s    = expf(ut_s - lse_t);
    const float pt_o    = expf(ut_o - lse_t);
    const float lt_cat  = ((cat == samp) ? ut_cc : ut_c) - lse_t;
    const float lt_s    = ut_s - lse_t;
    const float lt_o    = ut_o - lse_t;

    // ---- model posterior in linear domain ----
    // exp(evm_k) = A2*s_k + Cc  with A2 = exp(lc_p)/se  (t==0: 1/se, Cc=0)
    const float A2 = exp_lc / se;
    // Z = sum_k exp(evm_k + qx_k), closed form from se and s_samp
    const float Z = A2 * (Q_hit * s_samp + Q_miss * (se - s_samp))
                  + Cc * (Q_hit + 15.0f * Q_miss);
    const float lse_m = logf(Z);

    // ---- KL(true || model) and NLL ----
    float kl = 0.0f, nll = 0.0f;
    #pragma unroll
    for (int k = 0; k < KCLS; ++k) {
        float evm = logf(fmaf(x[k], A2, Cc));                 // 1 log per k
        float um  = evm + ((k == samp) ? qx_hit : qx_miss);
        float lm  = um - lse_m;                               // log_model_k
        float lt  = (k == cat) ? lt_cat : ((k == samp) ? lt_s : lt_o);
        float pt  = (k == cat) ? pt_cat : ((k == samp) ? pt_s : pt_o);
        kl  += pt * (lt - lm);
        nll -= ((k == cat) ? 1.0f : 1e-30f) * lm;
    }
    const float Lt = t0 ? nll : kl;

    // ---- exact f32 wave reduction via v_wmma, then f64 global atomic ----
    float wsum = wave_sum_wmma(Lt);
    if ((threadIdx.x & 31) == 0) atomicAdd(&acc[0], (double)wsum);
}

// ---------------------------------------------------------------------------
// Kernel 3: numeric MSE partials. Grid chosen so every thread does exactly
// NUM_N / (grid*block) iterations -> EXEC stays full for the WMMA reduce.
// ---------------------------------------------------------------------------
__global__ void __launch_bounds__(256)
md_num_kernel(const float* __restrict__ pred, const float* __restrict__ noise,
              double* __restrict__ acc) {
    const int stride = gridDim.x * blockDim.x;
    float s = 0.0f;
    for (int i = blockIdx.x * blockDim.x + threadIdx.x; i < NUM_N; i += stride) {
        float d = pred[i] - noise[i];
        s += d * d;
    }
    float wsum = wave_sum_wmma(s);
    if ((threadIdx.x & 31) == 0) atomicAdd(&acc[1], (double)wsum);
}

// ---------------------------------------------------------------------------
// Kernel 4: finalize. kl_prior is data-independent (one-hot) => closed form.
// ---------------------------------------------------------------------------
__global__ void md_final_kernel(const double* __restrict__ acc,
                                const float* __restrict__ lc,
                                const float* __restrict__ l1c,
                                float* __restrict__ out) {
    if (threadIdx.x != 0 || blockIdx.x != 0) return;
    const float LOG_EPS = logf(1e-30f);
    const float LOG_KF  = logf(16.0f);
    const float lcT  = lc[T_STEPS - 1];
    const float l1cT = l1c[T_STEPS - 1];
    const float hit  = lae(lcT, l1cT - LOG_KF);
    const float miss = lae(LOG_EPS + lcT, l1cT - LOG_KF);
    const float seg = expf(hit) * (hit + LOG_KF) + 15.0f * expf(miss) * (miss + LOG_KF);
    const double kl_prior = (double)(NCAT * LSEQ) * (double)seg;   // per batch
    const double loss_cat = acc[0] / (double)BATCH + kl_prior;
    const double loss_num = acc[1] / (double)NUM_N;
    out[0] = (float)(0.5 * loss_num + 0.5 * loss_cat);
}

extern "C" void kernel_launch(void* const* d_in, const int* in_sizes, int n_in,
                              void* d_out, int out_size, void* d_ws, size_t ws_size,
                              hipStream_t stream) {
    (void)in_sizes; (void)n_in; (void)out_size; (void)ws_size;
    const int*   cats      = (const int*)  d_in[0];
    const float* logits    = (const float*)d_in[1];
    const float* num_out   = (const float*)d_in[2];
    const float* noise_num = (const float*)d_in[3];
    const float* u_noise   = (const float*)d_in[4];
    const int*   t_arr     = (const int*)  d_in[5];
    float* out = (float*)d_out;

    // workspace: [2 x double accumulators][4 x 1000 f32 schedule buffers]
    double* acc = (double*)d_ws;
    float*  sched = (float*)((char*)d_ws + 2 * sizeof(double));
    float* la  = sched;
    float* l1a = sched + T_STEPS;
    float* lc  = sched + 2 * T_STEPS;
    float* l1c = sched + 3 * T_STEPS;

    md_sched_kernel<<<1, 32, 0, stream>>>(la, l1a, lc, l1c, acc);
    md_cat_kernel<<<BATCH * NCAT, LSEQ, 0, stream>>>(cats, logits, u_noise, t_arr,
                                                     la, l1a, lc, l1c, acc);
    md_num_kernel<<<128, 256, 0, stream>>>(num_out, noise_num, acc);
    md_final_kernel<<<1, 32, 0, stream>>>(acc, lc, l1c, out);
}